// CapsuleLayer_10746008175519
// MI455X (gfx1250) — compile-verified
//
#include <hip/hip_runtime.h>

// CapsuleLayer dynamic routing for MI455X (gfx1250, wave32).
// u_hat (335 MB) is never materialized: it is recomputed from W (10.5 MB,
// L2-resident) with chained V_WMMA_F32_16X16X4_F32.
//   caps_s:    c folded into the WMMA A operand (per-lane scalar, b = lm),
//              WMMA accumulates s directly through its C operand.
//   caps_bupd: transposed WMMA (swap A/B operands -> u_hat^T tile) turns the
//              dot over d into an in-lane FMA chain + one cross-half shuffle.
// Workspace: logits[B][I][J] (21 MB) + s[B][J][16] + v[B][J][16].

#define B_TOT 256
#define I_TOT 2048
#define K_DIM 8
#define J_CAP 10
#define D_DIM 16
#define EPSF  1e-7f

#define WAVES 4      // waves per block
#define ITILE 64     // i-columns per block (each wave strides by WAVES)

typedef float v2f __attribute__((ext_vector_type(2)));
typedef float v4f __attribute__((ext_vector_type(4)));
typedef float v8f __attribute__((ext_vector_type(8)));

__global__ void caps_zero(float* __restrict__ p, int n) {
  int t = blockIdx.x * blockDim.x + threadIdx.x;
  if (t < n) p[t] = 0.0f;
}

// ---------------------------------------------------------------------------
// s[b,j,:] += sum_i c[b,j,i] * (x[b,i,:] @ W[j,i]^T)
// grid = (I_TOT/ITILE, B_TOT/16), block = WAVES*32
//
// WMMA operands (V_WMMA_F32_16X16X4_F32, D[M,N] = A[M,K]B[K,N]+C):
//   A (16x4): lane holds row M=lm, v2f = x[b0+lm][i][k0..k0+1], k0 = 2*(lane/16)
//   B (4x16): lane holds col N=lm, v2f = W[j][i][lm][k0..k0+1]
//   D tile:   lane,VGPR v -> (b = b0+v+8*(lane/16), d = lm)
// c[b,j,i] multiplies A rows -> single per-lane scalar per j.
// ---------------------------------------------------------------------------
__global__ __launch_bounds__(WAVES * 32)
void caps_s(const float* __restrict__ x,       // [B][I][8]
            const float* __restrict__ W,       // [J][I][16][8]
            const float* __restrict__ logits,  // [B][I][J]
            float* __restrict__ s_buf,         // [B][J][16]
            int round0) {
  const int lane = threadIdx.x & 31;
  const int wave = threadIdx.x >> 5;
  const int half = lane >> 4;
  const int lm   = lane & 15;
  const int k0   = half * 2;
  const int b0     = blockIdx.y * 16;
  const int i_base = blockIdx.x * ITILE;

  v8f sacc[J_CAP];
#pragma unroll
  for (int j = 0; j < J_CAP; ++j) sacc[j] = (v8f){};

  for (int i = i_base + wave; i < i_base + ITILE; i += WAVES) {
    // per-lane coupling coefficients for row b = b0+lm (both halves duplicate)
    float cj[J_CAP];
    if (round0) {
#pragma unroll
      for (int j = 0; j < J_CAP; ++j) cj[j] = 0.1f;
    } else {
      const v2f* lp = (const v2f*)(logits + ((size_t)(b0 + lm) * I_TOT + i) * J_CAP);
      v2f l0 = lp[0], l1 = lp[1], l2 = lp[2], l3 = lp[3], l4 = lp[4];
      cj[0] = l0.x; cj[1] = l0.y; cj[2] = l1.x; cj[3] = l1.y; cj[4] = l2.x;
      cj[5] = l2.y; cj[6] = l3.x; cj[7] = l3.y; cj[8] = l4.x; cj[9] = l4.y;
      float m = cj[0];
#pragma unroll
      for (int j = 1; j < J_CAP; ++j) m = fmaxf(m, cj[j]);
      float ssum = 0.f;
#pragma unroll
      for (int j = 0; j < J_CAP; ++j) { cj[j] = __expf(cj[j] - m); ssum += cj[j]; }
      float inv = 1.0f / ssum;
#pragma unroll
      for (int j = 0; j < J_CAP; ++j) cj[j] *= inv;
    }

    const float* ap = x + ((size_t)(b0 + lm) * I_TOT + i) * K_DIM + k0;
    v2f a_lo = *(const v2f*)(ap);
    v2f a_hi = *(const v2f*)(ap + 4);

#pragma unroll
    for (int j = 0; j < J_CAP; ++j) {
      const float* wp = W + (((size_t)j * I_TOT + i) * D_DIM + lm) * K_DIM + k0;
      v2f b_lo = *(const v2f*)(wp);
      v2f b_hi = *(const v2f*)(wp + 4);
      v2f as_lo = a_lo * cj[j];   // fold c into A rows (per-lane splat)
      v2f as_hi = a_hi * cj[j];
      sacc[j] = __builtin_amdgcn_wmma_f32_16x16x4_f32(false, as_lo, false, b_lo,
                                                      (short)0, sacc[j], false, false);
      sacc[j] = __builtin_amdgcn_wmma_f32_16x16x4_f32(false, as_hi, false, b_hi,
                                                      (short)0, sacc[j], false, false);
    }
  }

  // flush partial s tile: element (v, lane) -> (b = b0+v+8*half, d = lm)
#pragma unroll
  for (int j = 0; j < J_CAP; ++j)
#pragma unroll
    for (int v = 0; v < 8; ++v)
      atomicAdd(&s_buf[((size_t)(b0 + v + 8 * half) * J_CAP + j) * D_DIM + lm],
                sacc[j][v]);
}

// ---------------------------------------------------------------------------
// squash: out[b,j,:] = s*|s|^2 / ((1+|s|^2) sqrt(|s|^2+eps)); one thread per (b,j)
// ---------------------------------------------------------------------------
__global__ void caps_squash(const float* __restrict__ s_buf,
                            float* __restrict__ out) {
  int t = blockIdx.x * blockDim.x + threadIdx.x;
  if (t >= B_TOT * J_CAP) return;
  const float* sp = s_buf + (size_t)t * D_DIM;
  float s2 = 0.f;
#pragma unroll
  for (int d = 0; d < D_DIM; ++d) s2 += sp[d] * sp[d];
  float scale = s2 / (1.0f + s2) * rsqrtf(s2 + EPSF);
#pragma unroll
  for (int d = 0; d < D_DIM; ++d) out[(size_t)t * D_DIM + d] = scale * sp[d];
}

// ---------------------------------------------------------------------------
// logits[b,i,j] (+)= dot_d(v[b,j,:], u_hat[b,j,i,:])
// Transposed WMMA: u^T[d,b] tile by swapping A/B operands (same loads!).
//   D tile: lane,VGPR v -> (d = v+8*half, b = b0+lm)
// Dot over d = 8-term in-lane FMA chain + one shfl_xor(16) across halves.
// ---------------------------------------------------------------------------
__global__ __launch_bounds__(WAVES * 32)
void caps_bupd(const float* __restrict__ x,      // [B][I][8]
               const float* __restrict__ W,      // [J][I][16][8]
               const float* __restrict__ v_buf,  // [B][J][16]
               float* __restrict__ logits,       // [B][I][J]
               int round0) {
  const int lane = threadIdx.x & 31;
  const int wave = threadIdx.x >> 5;
  const int half = lane >> 4;
  const int lm   = lane & 15;
  const int k0   = half * 2;
  const int b0     = blockIdx.y * 16;
  const int i_base = blockIdx.x * ITILE;

  // i-invariant: vt[j][v] = v_buf[b0+lm][j][d = 8*half+v]  (contiguous, aligned)
  v4f vt0[J_CAP], vt1[J_CAP];
#pragma unroll
  for (int j = 0; j < J_CAP; ++j) {
    const float* vp = v_buf + ((size_t)(b0 + lm) * J_CAP + j) * D_DIM + 8 * half;
    vt0[j] = *(const v4f*)(vp);
    vt1[j] = *(const v4f*)(vp + 4);
  }

  for (int i = i_base + wave; i < i_base + ITILE; i += WAVES) {
    const float* ap = x + ((size_t)(b0 + lm) * I_TOT + i) * K_DIM + k0;
    v2f a_lo = *(const v2f*)(ap);
    v2f a_hi = *(const v2f*)(ap + 4);

    float areg[J_CAP];
#pragma unroll
    for (int j = 0; j < J_CAP; ++j) {
      const float* wp = W + (((size_t)j * I_TOT + i) * D_DIM + lm) * K_DIM + k0;
      v2f b_lo = *(const v2f*)(wp);
      v2f b_hi = *(const v2f*)(wp + 4);
      // swapped operands: A = W rows (M=d), B = x^T cols (N=b) -> u^T tile
      v8f u = {};
      u = __builtin_amdgcn_wmma_f32_16x16x4_f32(false, b_lo, false, a_lo,
                                                (short)0, u, false, false);
      u = __builtin_amdgcn_wmma_f32_16x16x4_f32(false, b_hi, false, a_hi,
                                                (short)0, u, false, false);
      // p = sum_{d in this half} v[b,j,d] * u^T[d,b]
      float p = u[0] * vt0[j].x;
      p = fmaf(u[1], vt0[j].y, p);
      p = fmaf(u[2], vt0[j].z, p);
      p = fmaf(u[3], vt0[j].w, p);
      p = fmaf(u[4], vt1[j].x, p);
      p = fmaf(u[5], vt1[j].y, p);
      p = fmaf(u[6], vt1[j].z, p);
      p = fmaf(u[7], vt1[j].w, p);
      areg[j] = p + __shfl_xor(p, 16);  // combine halves: full dot over d
    }

    // every lane now holds a[b0+lm, i, 0..9]; half 0 writes j=0..4, half 1 j=5..9
    size_t base = ((size_t)(b0 + lm) * I_TOT + i) * J_CAP + 5 * half;
#pragma unroll
    for (int q = 0; q < 5; ++q) {
      float add = half ? areg[5 + q] : areg[q];
      logits[base + q] = round0 ? add : (logits[base + q] + add);
    }
  }
}

// ---------------------------------------------------------------------------
extern "C" void kernel_launch(void* const* d_in, const int* in_sizes, int n_in,
                              void* d_out, int out_size, void* d_ws, size_t ws_size,
                              hipStream_t stream) {
  (void)in_sizes; (void)n_in; (void)out_size; (void)ws_size;
  const float* x = (const float*)d_in[0];  // [256][2048][8]
  const float* W = (const float*)d_in[1];  // [10][2048][16][8]
  float* out = (float*)d_out;              // [256][10][16]

  // workspace: logits (21 MB) | s (160 KB) | v (160 KB)
  float* logits = (float*)d_ws;
  float* s_buf  = logits + (size_t)B_TOT * I_TOT * J_CAP;
  float* v_buf  = s_buf + (size_t)B_TOT * J_CAP * D_DIM;

  const int s_elems = B_TOT * J_CAP * D_DIM;  // 40960
  dim3 gridS(I_TOT / ITILE, B_TOT / 16);      // (32, 16)
  dim3 blockS(WAVES * 32);

  for (int r = 0; r < 3; ++r) {
    caps_zero<<<(s_elems + 255) / 256, 256, 0, stream>>>(s_buf, s_elems);
    caps_s<<<gridS, blockS, 0, stream>>>(x, W, logits, s_buf, r == 0 ? 1 : 0);
    float* vout = (r == 2) ? out : v_buf;
    caps_squash<<<(B_TOT * J_CAP + 255) / 256, 256, 0, stream>>>(s_buf, vout);
    if (r < 2)
      caps_bupd<<<gridS, blockS, 0, stream>>>(x, W, v_buf, logits, r == 0 ? 1 : 0);
  }
}